// GuidedAttentionLoss_8572754723318
// MI455X (gfx1250) — compile-verified
//
#include <hip/hip_runtime.h>
#include <hip/hip_bf16.h>

typedef __attribute__((ext_vector_type(4))) float v4f;
typedef __attribute__((ext_vector_type(2))) float v2f;
typedef __attribute__((ext_vector_type(8))) float v8f;

#define GAL_T_OUT 2000
#define GAL_T_IN  512
#define GAL_BATCH 64
#define GAL_NBLK  2000
#define GAL_WAVES 8
#define GAL_ITERS 8      // rows per wave; 2000 blocks * 8 waves * 8 = 128000 = B*T_out exactly
// SIGMA = 0.4 -> 1/(2*sigma^2) = 3.125

__global__ __launch_bounds__(256) void gal_partial_kernel(
    const float* __restrict__ A,
    const int*   __restrict__ in_len,
    const int*   __restrict__ out_len,
    float*       __restrict__ partial)
{
    const int lane = threadIdx.x & 31;
    const int wave = threadIdx.x >> 5;

    float s = 0.0f;
    const int rowbase = blockIdx.x * (GAL_WAVES * GAL_ITERS) + wave * GAL_ITERS;

#pragma unroll
    for (int it = 0; it < GAL_ITERS; ++it) {
        const int row = rowbase + it;          // row = b*T_OUT + i
        const int b   = row / GAL_T_OUT;
        const int i   = row - b * GAL_T_OUT;
        const int Ti  = in_len[b];
        const int To  = out_len[b];
        if (i < To) {                          // uniform per wave (row is wave-uniform)
            const float ratio = (float)To / (float)Ti;
            const float fi    = (float)i;
            const float* rp   = A + (size_t)row * GAL_T_IN;
#pragma unroll
            for (int ch = 0; ch < 4; ++ch) {
                const int j0 = ch * 128 + lane * 4;       // coalesced 512B per wave-load
                v4f v = __builtin_nontemporal_load((const v4f*)(rp + j0));
#pragma unroll
                for (int e = 0; e < 4; ++e) {
                    const int   j = j0 + e;
                    const float d = fi - (float)j * ratio;
                    const float w = 1.0f - __expf(d * d * -3.125f);
                    s += (j < Ti) ? v[e] * w : 0.0f;      // branchless mask, EXEC stays full
                }
            }
        }
    }

    // ---- wave32 cross-lane sum via V_WMMA_F32_16X16X4_F32 ----
    // A layout (16x4 f32): lanes 0-15 -> A[lane,0]=a.x, A[lane,1]=a.y
    //                      lanes 16-31 -> A[lane-16,2]=a.x, A[lane-16,3]=a.y
    // With B = all-ones: D[m,n] = s_m + s_(m+16). Per-lane sum of the 8 D VGPRs
    // gives half-column sums; adding the X16-swapped half yields the wave total.
    v2f a;  a.x  = s;    a.y  = 0.0f;
    v2f bm; bm.x = 1.0f; bm.y = 1.0f;
    v8f c = {};
    v8f dm = __builtin_amdgcn_wmma_f32_16x16x4_f32(
        /*neg_a=*/false, a, /*neg_b=*/false, bm,
        /*c_mod=*/(short)0, c, /*reuse_a=*/false, /*reuse_b=*/false);
    float t = dm[0] + dm[1] + dm[2] + dm[3] + dm[4] + dm[5] + dm[6] + dm[7];
    float wave_tot = t + __shfl_xor(t, 16, 32);

    __shared__ float wsum[GAL_WAVES];
    if (lane == 0) wsum[wave] = wave_tot;
    __syncthreads();
    if (threadIdx.x == 0) {
        float bsum = 0.0f;
#pragma unroll
        for (int wv = 0; wv < GAL_WAVES; ++wv) bsum += wsum[wv];
        partial[blockIdx.x] = bsum;            // no atomics -> deterministic
    }
}

__global__ __launch_bounds__(256) void gal_final_kernel(
    const float* __restrict__ partial, float* __restrict__ out,
    int n, float scale)
{
    __shared__ float red[256];
    float s = 0.0f;
    for (int idx = threadIdx.x; idx < n; idx += 256) s += partial[idx];
    red[threadIdx.x] = s;
    __syncthreads();
    for (int off = 128; off > 0; off >>= 1) {
        if (threadIdx.x < off) red[threadIdx.x] += red[threadIdx.x + off];
        __syncthreads();
    }
    if (threadIdx.x == 0) out[0] = red[0] * scale;
}

extern "C" void kernel_launch(void* const* d_in, const int* in_sizes, int n_in,
                              void* d_out, int out_size, void* d_ws, size_t ws_size,
                              hipStream_t stream)
{
    const float* A  = (const float*)d_in[0];   // alignments [64,2000,512] f32
    const int*   il = (const int*)d_in[1];     // input_lengths [64] i32
    const int*   ol = (const int*)d_in[2];     // output_lengths [64] i32
    float* out     = (float*)d_out;            // scalar f32
    float* partial = (float*)d_ws;             // 2000 floats of scratch

    gal_partial_kernel<<<GAL_NBLK, 256, 0, stream>>>(A, il, ol, partial);
    gal_final_kernel<<<1, 256, 0, stream>>>(partial, out, GAL_NBLK, 1.0f / GAL_BATCH);
}